// PSAGE_41351945126325
// MI455X (gfx1250) — compile-verified
//
#include <hip/hip_runtime.h>
#include <hip/hip_bf16.h>

// ---------------------------------------------------------------------------
// CDNA5 (gfx1250) implementation of the 3-layer GraphSAGE + MLP head.
// GEMMs use V_WMMA_F32_16X16X4_F32 (f32 in/out, matches reference precision).
// Per-wave macro-tile: 32 rows x 64 cols (2x4 WMMA subtiles), so each K-step
// issues 16 WMMAs on 12 b64 loads (B panels reused 2x, A rows reused 4x).
// Scatter/mean aggregation = coalesced f32 global atomics (resolve in the
// 192MB L2: x and agg both fit).
// ---------------------------------------------------------------------------

#define HDIM 128

typedef float v2f __attribute__((ext_vector_type(2)));
typedef float v8f __attribute__((ext_vector_type(8)));

static __device__ __forceinline__ v8f wmma_f32_4(v2f a, v2f b, v8f c) {
    // 8-arg pattern: (neg_a, A, neg_b, B, c_mod, C, reuse_a, reuse_b)
    return __builtin_amdgcn_wmma_f32_16x16x4_f32(false, a, false, b, (short)0, c,
                                                 false, false);
}

// ---------------------------------------------------------------- utilities
__global__ void zero_f32(float* __restrict__ p, long long n) {
    long long i = (long long)blockIdx.x * blockDim.x + threadIdx.x;
    if (i < n) p[i] = 0.0f;
}
__global__ void zero_i32(int* __restrict__ p, long long n) {
    long long i = (long long)blockIdx.x * blockDim.x + threadIdx.x;
    if (i < n) p[i] = 0;
}

// ------------------------------------------------------------ degree + init
__global__ void deg_count(const int* __restrict__ dst, int* __restrict__ deg, int E) {
    int i = blockIdx.x * blockDim.x + threadIdx.x;
    if (i < E) atomicAdd(&deg[dst[i]], 1);
}
__global__ void inv_deg_k(const int* __restrict__ deg, float* __restrict__ invdeg, int Nn) {
    int i = blockIdx.x * blockDim.x + threadIdx.x;
    if (i < Nn) invdeg[i] = 1.0f / fmaxf((float)deg[i], 1.0f);
}

// x[b,n,:] = z_table[ (z[n]==b+1) ? 1 : 0 ]
__global__ void init_x(const int* __restrict__ z, const float* __restrict__ z_table,
                       float* __restrict__ X, int Nn) {
    long long idx = (long long)blockIdx.x * blockDim.x + threadIdx.x;
    long long total = 2LL * Nn * HDIM;
    if (idx >= total) return;
    int hh = (int)(idx % HDIM);
    long long r = idx / HDIM;
    int n = (int)(r % Nn);
    int b = (int)(r / Nn);
    int lbl = (z[n] == b + 1) ? 1 : 0;
    X[idx] = z_table[lbl * HDIM + hh];
}

// --------------------------------------------------------------- scatter add
// agg[b,dst,:] += x[b,src,:]  for both batch slices; lanes cover consecutive h
__global__ void scatter_add(const float* __restrict__ X, const int* __restrict__ srcE,
                            const int* __restrict__ dstE, float* __restrict__ AGG,
                            int E, int Nn) {
    long long idx = (long long)blockIdx.x * blockDim.x + threadIdx.x;
    long long total = (long long)E * HDIM;
    if (idx >= total) return;
    int e  = (int)(idx / HDIM);
    int hh = (int)(idx % HDIM);
    int s = srcE[e], d = dstE[e];
    atomicAdd(&AGG[(size_t)d * HDIM + hh],        X[(size_t)s * HDIM + hh]);
    atomicAdd(&AGG[((size_t)Nn + d) * HDIM + hh], X[((size_t)(Nn + s)) * HDIM + hh]);
}

// agg[r,:] *= invdeg[r % Nn]   (hoists the mean-divide out of the WMMA loop)
__global__ void scale_agg(float* __restrict__ AGG, const float* __restrict__ invdeg,
                          long long total, int Nn) {
    long long idx = (long long)blockIdx.x * blockDim.x + threadIdx.x;
    if (idx >= total) return;
    int n = (int)((idx / HDIM) % Nn);
    AGG[idx] *= invdeg[n];
}

// ----------------------------------------------------------- fused SAGE GEMM
// Y = maybe_relu( AGGs @ Wl.T + bl + X @ Wr.T ),  rows R = 2*Nn
// one wave -> 32x64 macro-tile = 2x4 grid of 16x16 WMMA tiles.
#define RT 2
#define CT 4
__global__ void sage_gemm(const float* __restrict__ X, const float* __restrict__ AGG,
                          const float* __restrict__ Wl, const float* __restrict__ bl,
                          const float* __restrict__ Wr,
                          float* __restrict__ Y, int R, int doRelu) {
    const int wavesPerBlock = blockDim.x >> 5;
    int wave = threadIdx.x >> 5;
    int lane = threadIdx.x & 31;
    int tile = blockIdx.x * wavesPerBlock + wave;
    int rowMacros = (R + 16 * RT - 1) / (16 * RT);
    int colMacros = HDIM / (16 * CT);
    int T = rowMacros * colMacros;
    if (tile >= T) return;                 // wave-uniform exit (EXEC stays full)

    int rm = tile / colMacros;
    int cm = tile % colMacros;
    int g = lane >> 4;                     // half-wave group
    int m = lane & 15;
    int row0 = rm * 16 * RT;
    int col0 = cm * 16 * CT;

    v8f acc[RT][CT];
#pragma unroll
    for (int cc = 0; cc < CT; ++cc) {
        float blv = bl[col0 + cc * 16 + m];
#pragma unroll
        for (int rr = 0; rr < RT; ++rr)
#pragma unroll
            for (int v = 0; v < 8; ++v) acc[rr][cc][v] = blv;
    }

    const float* xr[RT];
    const float* ar[RT];
#pragma unroll
    for (int rr = 0; rr < RT; ++rr) {
        int r = row0 + rr * 16 + m;        // A row this lane feeds
        if (r >= R) r = R - 1;             // clamp (R multiple of 32 here)
        xr[rr] = X   + (size_t)r * HDIM;
        ar[rr] = AGG + (size_t)r * HDIM;
    }
    const float* wlr[CT];
    const float* wrr[CT];
#pragma unroll
    for (int cc = 0; cc < CT; ++cc) {
        int n = col0 + cc * 16 + m;        // B col this lane feeds (B[k][n]=W[n][k])
        wlr[cc] = Wl + (size_t)n * HDIM;
        wrr[cc] = Wr + (size_t)n * HDIM;
    }

#pragma unroll 2
    for (int k = 0; k < HDIM; k += 4) {
        int ko = k + 2 * g;                // per-lane K pair (K = v + 2g)
        v2f aa[RT], ax[RT], bwl[CT], bwr[CT];
#pragma unroll
        for (int rr = 0; rr < RT; ++rr) {
            aa[rr] = *(const v2f*)(ar[rr] + ko);
            ax[rr] = *(const v2f*)(xr[rr] + ko);
        }
#pragma unroll
        for (int cc = 0; cc < CT; ++cc) {
            bwl[cc] = *(const v2f*)(wlr[cc] + ko);
            bwr[cc] = *(const v2f*)(wrr[cc] + ko);
        }
#pragma unroll
        for (int rr = 0; rr < RT; ++rr)
#pragma unroll
            for (int cc = 0; cc < CT; ++cc) {
                acc[rr][cc] = wmma_f32_4(aa[rr], bwl[cc], acc[rr][cc]);
                acc[rr][cc] = wmma_f32_4(ax[rr], bwr[cc], acc[rr][cc]);
            }
    }

#pragma unroll
    for (int rr = 0; rr < RT; ++rr)
#pragma unroll
        for (int cc = 0; cc < CT; ++cc)
#pragma unroll
            for (int v = 0; v < 8; ++v) {
                int r = row0 + rr * 16 + v + 8 * g;   // D row for this lane/VGPR
                if (r < R) {
                    float val = acc[rr][cc][v];
                    if (doRelu) val = fmaxf(val, 0.0f);
                    Y[(size_t)r * HDIM + col0 + cc * 16 + m] = val;
                }
            }
}

// ----------------------------------------------------- h = sum_b relu(x@W0.T+b0)
// Same macro-tiling; batch-sum fused via f32 atomics into h[n].
__global__ void head_gemm(const float* __restrict__ X, const float* __restrict__ W0,
                          const float* __restrict__ b0, float* __restrict__ Hout,
                          int R, int Nn) {
    const int wavesPerBlock = blockDim.x >> 5;
    int wave = threadIdx.x >> 5;
    int lane = threadIdx.x & 31;
    int tile = blockIdx.x * wavesPerBlock + wave;
    int rowMacros = (R + 16 * RT - 1) / (16 * RT);
    int colMacros = HDIM / (16 * CT);
    int T = rowMacros * colMacros;
    if (tile >= T) return;

    int rm = tile / colMacros;
    int cm = tile % colMacros;
    int g = lane >> 4;
    int m = lane & 15;
    int row0 = rm * 16 * RT;
    int col0 = cm * 16 * CT;

    v8f acc[RT][CT];
#pragma unroll
    for (int cc = 0; cc < CT; ++cc) {
        float b0v = b0[col0 + cc * 16 + m];
#pragma unroll
        for (int rr = 0; rr < RT; ++rr)
#pragma unroll
            for (int v = 0; v < 8; ++v) acc[rr][cc][v] = b0v;
    }

    const float* xr[RT];
#pragma unroll
    for (int rr = 0; rr < RT; ++rr) {
        int r = row0 + rr * 16 + m;
        if (r >= R) r = R - 1;
        xr[rr] = X + (size_t)r * HDIM;
    }
    const float* w0r[CT];
#pragma unroll
    for (int cc = 0; cc < CT; ++cc)
        w0r[cc] = W0 + (size_t)(col0 + cc * 16 + m) * HDIM;

#pragma unroll 2
    for (int k = 0; k < HDIM; k += 4) {
        int ko = k + 2 * g;
        v2f ax[RT], bw[CT];
#pragma unroll
        for (int rr = 0; rr < RT; ++rr) ax[rr] = *(const v2f*)(xr[rr] + ko);
#pragma unroll
        for (int cc = 0; cc < CT; ++cc) bw[cc] = *(const v2f*)(w0r[cc] + ko);
#pragma unroll
        for (int rr = 0; rr < RT; ++rr)
#pragma unroll
            for (int cc = 0; cc < CT; ++cc)
                acc[rr][cc] = wmma_f32_4(ax[rr], bw[cc], acc[rr][cc]);
    }

#pragma unroll
    for (int rr = 0; rr < RT; ++rr)
#pragma unroll
        for (int cc = 0; cc < CT; ++cc)
#pragma unroll
            for (int v = 0; v < 8; ++v) {
                int r = row0 + rr * 16 + v + 8 * g;
                if (r < R) {
                    int n = r % Nn;                 // fold batch dim
                    atomicAdd(&Hout[(size_t)n * HDIM + col0 + cc * 16 + m],
                              fmaxf(acc[rr][cc][v], 0.0f));
                }
            }
}

// ----------------------------------------------------- stable index compaction
__global__ void count_flags(const int* __restrict__ z, int Nn,
                            int* __restrict__ cnt1, int* __restrict__ cnt2) {
    __shared__ int c1, c2;
    if (threadIdx.x == 0) { c1 = 0; c2 = 0; }
    __syncthreads();
    int i = blockIdx.x * blockDim.x + threadIdx.x;
    if (i < Nn) {
        int zi = z[i];
        if (zi == 1) atomicAdd(&c1, 1);
        else if (zi == 2) atomicAdd(&c2, 1);
    }
    __syncthreads();
    if (threadIdx.x == 0) { cnt1[blockIdx.x] = c1; cnt2[blockIdx.x] = c2; }
}

__global__ void scan_blocks(const int* __restrict__ cnt1, const int* __restrict__ cnt2,
                            int* __restrict__ off1, int* __restrict__ off2, int nb) {
    if (blockIdx.x == 0 && threadIdx.x == 0) {
        int s1 = 0, s2 = 0;
        for (int i = 0; i < nb; ++i) {
            off1[i] = s1; s1 += cnt1[i];
            off2[i] = s2; s2 += cnt2[i];
        }
    }
}

__global__ void scatter_idx(const int* __restrict__ z, int Nn,
                            const int* __restrict__ off1, const int* __restrict__ off2,
                            int* __restrict__ sidx, int* __restrict__ didx, int P) {
    __shared__ int flags[256];
    int tid = threadIdx.x;
    int i = blockIdx.x * blockDim.x + tid;
    int f1 = 0, f2 = 0;
    if (i < Nn) { int zi = z[i]; f1 = (zi == 1); f2 = (zi == 2); }
    flags[tid] = f1 | (f2 << 16);
    __syncthreads();
    // Hillis-Steele inclusive scan over packed counts
    for (int s = 1; s < 256; s <<= 1) {
        int t = (tid >= s) ? flags[tid - s] : 0;
        __syncthreads();
        flags[tid] += t;
        __syncthreads();
    }
    int inc = flags[tid];
    int rank1 = (inc & 0xFFFF) - f1;
    int rank2 = (inc >> 16) - f2;
    if (f1) { int pos = off1[blockIdx.x] + rank1; if (pos < P) sidx[pos] = i; }
    if (f2) { int pos = off2[blockIdx.x] + rank2; if (pos < P) didx[pos] = i; }
}

// -------------------------------------------------------------- pair product
__global__ void pair_mul(const float* __restrict__ Hbuf, const int* __restrict__ sidx,
                         const int* __restrict__ didx, float* __restrict__ Pmat, int P) {
    long long idx = (long long)blockIdx.x * blockDim.x + threadIdx.x;
    long long total = (long long)P * HDIM;
    if (idx >= total) return;
    int k  = (int)(idx / HDIM);
    int hh = (int)(idx % HDIM);
    Pmat[idx] = Hbuf[(size_t)sidx[k] * HDIM + hh] * Hbuf[(size_t)didx[k] * HDIM + hh];
}

// ------------------------------------------------------------------ MLP head
// out[k] = W2 . relu(W1 @ p[k] + b1) + b2      (tiny: 1024 x 128 x 128)
__global__ void mlp_out(const float* __restrict__ Pmat, const float* __restrict__ W1,
                        const float* __restrict__ b1, const float* __restrict__ W2,
                        const float* __restrict__ b2, float* __restrict__ out) {
    __shared__ float prow[HDIM];
    __shared__ float red[HDIM];
    int k = blockIdx.x;
    int j = threadIdx.x;               // 128 threads
    prow[j] = Pmat[(size_t)k * HDIM + j];
    __syncthreads();
    float acc = b1[j];
    const float* wr = W1 + (size_t)j * HDIM;
#pragma unroll 8
    for (int i = 0; i < HDIM; ++i) acc += wr[i] * prow[i];
    red[j] = fmaxf(acc, 0.0f) * W2[j];
    __syncthreads();
    for (int s = 64; s > 0; s >>= 1) {
        if (j < s) red[j] += red[j + s];
        __syncthreads();
    }
    if (j == 0) out[k] = red[0] + b2[0];
}

// ---------------------------------------------------------------------------
extern "C" void kernel_launch(void* const* d_in, const int* in_sizes, int n_in,
                              void* d_out, int out_size, void* d_ws, size_t ws_size,
                              hipStream_t stream) {
    (void)n_in; (void)ws_size;
    const int H  = HDIM;
    const int Nn = in_sizes[0];
    const int E  = in_sizes[1] / 2;
    const int L  = in_sizes[4] / (H * H);
    const int P  = out_size;

    const int*   z       = (const int*)d_in[0];
    const int*   ei      = (const int*)d_in[1];
    const int*   srcE    = ei;
    const int*   dstE    = ei + E;
    const float* z_table = (const float*)d_in[3];
    const float* Wl      = (const float*)d_in[4];
    const float* blp     = (const float*)d_in[5];
    const float* Wr      = (const float*)d_in[6];
    const float* W0      = (const float*)d_in[7];
    const float* b0      = (const float*)d_in[8];
    const float* W1      = (const float*)d_in[9];
    const float* b1      = (const float*)d_in[10];
    const float* W2      = (const float*)d_in[11];
    const float* b2      = (const float*)d_in[12];
    float*       out     = (float*)d_out;

    // ---- carve workspace
    const size_t R = 2 * (size_t)Nn;
    char* base = (char*)d_ws;
    float* xA     = (float*)base; base += R * H * sizeof(float);
    float* xB     = (float*)base; base += R * H * sizeof(float);
    float* agg    = (float*)base; base += R * H * sizeof(float);
    float* hbuf   = (float*)base; base += (size_t)Nn * H * sizeof(float);
    float* pbuf   = (float*)base; base += (size_t)P * H * sizeof(float);
    float* invdeg = (float*)base; base += (size_t)Nn * sizeof(float);
    int*   deg    = (int*)base;   base += (size_t)Nn * sizeof(int);
    const int nb  = (Nn + 255) / 256;
    int* cnt1 = (int*)base; base += nb * sizeof(int);
    int* cnt2 = (int*)base; base += nb * sizeof(int);
    int* off1 = (int*)base; base += nb * sizeof(int);
    int* off2 = (int*)base; base += nb * sizeof(int);
    int* sidx = (int*)base; base += P * sizeof(int);
    int* didx = (int*)base; base += P * sizeof(int);

    const int TB = 256;
    auto blocks = [](long long n, int tb) { return (unsigned)((n + tb - 1) / tb); };

    // ---- degree + inverse degree
    zero_i32<<<blocks(Nn, TB), TB, 0, stream>>>(deg, Nn);
    deg_count<<<blocks(E, TB), TB, 0, stream>>>(dstE, deg, E);
    inv_deg_k<<<blocks(Nn, TB), TB, 0, stream>>>(deg, invdeg, Nn);

    // ---- feature init
    init_x<<<blocks(2LL * Nn * H, TB), TB, 0, stream>>>(z, z_table, xA, Nn);

    // ---- stable compaction of z==1 / z==2 indices
    count_flags<<<nb, 256, 0, stream>>>(z, Nn, cnt1, cnt2);
    scan_blocks<<<1, 1, 0, stream>>>(cnt1, cnt2, off1, off2, nb);
    scatter_idx<<<nb, 256, 0, stream>>>(z, Nn, off1, off2, sidx, didx, P);

    // ---- SAGE layers
    const int rowMacros = (int)((R + 16 * RT - 1) / (16 * RT));
    const int colMacros = H / (16 * CT);
    const int tiles     = rowMacros * colMacros;
    const unsigned gemmBlocks = blocks(tiles, TB / 32);

    float* xcur = xA;
    float* xnxt = xB;
    for (int l = 0; l < L; ++l) {
        zero_f32<<<blocks((long long)R * H, TB), TB, 0, stream>>>(agg, (long long)R * H);
        scatter_add<<<blocks((long long)E * H, TB), TB, 0, stream>>>(xcur, srcE, dstE, agg, E, Nn);
        scale_agg<<<blocks((long long)R * H, TB), TB, 0, stream>>>(agg, invdeg, (long long)R * H, Nn);
        sage_gemm<<<gemmBlocks, TB, 0, stream>>>(xcur, agg,
                                                 Wl + (size_t)l * H * H, blp + (size_t)l * H,
                                                 Wr + (size_t)l * H * H,
                                                 xnxt, (int)R, (l < L - 1) ? 1 : 0);
        float* t = xcur; xcur = xnxt; xnxt = t;
    }

    // ---- h = sum_b relu(x @ W0.T + b0)
    zero_f32<<<blocks((long long)Nn * H, TB), TB, 0, stream>>>(hbuf, (long long)Nn * H);
    head_gemm<<<gemmBlocks, TB, 0, stream>>>(xcur, W0, b0, hbuf, (int)R, Nn);

    // ---- pair product + MLP head
    pair_mul<<<blocks((long long)P * H, TB), TB, 0, stream>>>(hbuf, sidx, didx, pbuf, P);
    mlp_out<<<P, HDIM, 0, stream>>>(pbuf, W1, b1, W2, b2, out);
}